// PyGHGNNConv_55602646614711
// MI455X (gfx1250) — compile-verified
//
#include <hip/hip_runtime.h>

typedef __attribute__((ext_vector_type(2))) float v2f;
typedef __attribute__((ext_vector_type(8))) float v8f;
typedef __attribute__((ext_vector_type(4))) float v4f;

#define N_NODES 100000
#define N_EDGES 20000
#define NNZ     1000000
#define IN_CH   128
#define OUT_DIM 64
#define BROW    160   // padded LDS row stride (floats) -> odd rows start at bank 32

// -------- zero a float buffer (float4 stores, grid-stride) ----------------
__global__ void __launch_bounds__(256) zero4_kernel(v4f* __restrict__ p, int n4) {
    int i = blockIdx.x * blockDim.x + threadIdx.x;
    int stride = gridDim.x * blockDim.x;
    v4f z = {};
    for (; i < n4; i += stride) p[i] = z;
}

// -------- Xl = X @ W_lin via V_WMMA_F32_16X16X4_F32 -----------------------
// One wave per 16x64 output row-block (4 accumulators). 6250 waves total;
// 10 waves (320 threads) per block -> 625 blocks, no tail.
// W_lin staged in LDS in WMMA-fragment-native layout:
//   (k,n) -> lW2[(k>>1)*BROW + n*2 + (k&1)]
// so each B fragment is a single aligned ds_load_b64, bank-conflict-free.
__global__ void __launch_bounds__(320)
gemm_wmma_f32(const float* __restrict__ X, const float* __restrict__ W,
              float* __restrict__ Xl) {
    __shared__ float lW2[(IN_CH / 2) * BROW];   // 40 KB of the 320 KB WGP LDS
    for (int i = threadIdx.x; i < IN_CH * OUT_DIM; i += 320) {
        const int k = i >> 6;          // 0..127
        const int n = i & 63;          // 0..63
        lW2[(k >> 1) * BROW + n * 2 + (k & 1)] = W[i];
    }
    __syncthreads();

    const int wave  = threadIdx.x >> 5;
    const int lane  = threadIdx.x & 31;
    const int mTile = blockIdx.x * 10 + wave;   // 0..6249
    const int lrow  = lane & 15;
    const int half  = lane >> 4;
    const int mBase = mTile * 16;

    const float* __restrict__ Arow = X + (size_t)(mBase + lrow) * IN_CH;

    v8f c0 = {}, c1 = {}, c2 = {}, c3 = {};
    #pragma unroll
    for (int kb = 0; kb < IN_CH / 4; ++kb) {
        const int k0 = kb * 4 + half * 2;       // K pair for this half-wave
        // A 16x4 f32 layout: lanes 0-15 hold K=0/1, lanes 16-31 hold K=2/3
        v2f a = *(const v2f*)(Arow + k0);
        // B fragments: one ds_load_b64 each, straight into the WMMA reg pair.
        // Lanes 0-15 read row (2kb+0) -> banks 0-31; lanes 16-31 read row
        // (2kb+1), offset 640B -> banks 32-63: conflict-free.
        const float* __restrict__ bp = lW2 + (size_t)(kb * 2 + half) * BROW + lrow * 2;
        v2f b0 = *(const v2f*)(bp     );
        v2f b1 = *(const v2f*)(bp + 32);
        v2f b2 = *(const v2f*)(bp + 64);
        v2f b3 = *(const v2f*)(bp + 96);
        c0 = __builtin_amdgcn_wmma_f32_16x16x4_f32(false, a, false, b0, (short)0, c0, false, false);
        c1 = __builtin_amdgcn_wmma_f32_16x16x4_f32(false, a, false, b1, (short)0, c1, false, false);
        c2 = __builtin_amdgcn_wmma_f32_16x16x4_f32(false, a, false, b2, (short)0, c2, false, false);
        c3 = __builtin_amdgcn_wmma_f32_16x16x4_f32(false, a, false, b3, (short)0, c3, false, false);
    }

    // D layout: VGPR v -> M = v + 8*half, N = lrow (+ 16*nTile)
    float* __restrict__ Dcol = Xl + (size_t)lrow;
    #pragma unroll
    for (int v = 0; v < 8; ++v) {
        const size_t m = (size_t)(mBase + v + 8 * half) * OUT_DIM;
        Dcol[m     ] = c0[v];
        Dcol[m + 16] = c1[v];
        Dcol[m + 32] = c2[v];
        Dcol[m + 48] = c3[v];
    }
}

// -------- phase 1 scatter: Xe[e] += Xl[v]  (one wave per incidence) -------
__global__ void __launch_bounds__(256)
scatter_v2e(const float* __restrict__ Xl, const int* __restrict__ vertex,
            const int* __restrict__ edges, float* __restrict__ Xe) {
    const int lane  = threadIdx.x & 31;
    const int gwave = blockIdx.x * (blockDim.x >> 5) + (threadIdx.x >> 5);
    const int nwave = gridDim.x * (blockDim.x >> 5);
    for (int i = gwave; i < NNZ; i += nwave) {
        const int v = vertex[i];
        const int e = edges[i];
        // 32 lanes x float2 = the full 64-float row, 256B coalesced
        v2f x = *(const v2f*)(Xl + (size_t)v * OUT_DIM + lane * 2);
        float* dst = Xe + (size_t)e * OUT_DIM + lane * 2;
        atomicAdd(dst,     x.x);
        atomicAdd(dst + 1, x.y);
    }
}

// -------- phase 2 scatter: out[v] += Xe[e] * (degE[e]*W_edge[e]) ----------
// Edge scaling fused into the gather: (Xe*s)[e] == Xe[e]*s[e].
__global__ void __launch_bounds__(256)
scatter_e2v(const float* __restrict__ Xe, const int* __restrict__ vertex,
            const int* __restrict__ edges, const float* __restrict__ degE,
            const float* __restrict__ Wedge, float* __restrict__ out) {
    const int lane  = threadIdx.x & 31;
    const int gwave = blockIdx.x * (blockDim.x >> 5) + (threadIdx.x >> 5);
    const int nwave = gridDim.x * (blockDim.x >> 5);
    for (int i = gwave; i < NNZ; i += nwave) {
        const int v = vertex[i];
        const int e = edges[i];
        const float s = degE[e] * Wedge[e];
        v2f x = *(const v2f*)(Xe + (size_t)e * OUT_DIM + lane * 2);
        float* dst = out + (size_t)v * OUT_DIM + lane * 2;
        atomicAdd(dst,     x.x * s);
        atomicAdd(dst + 1, x.y * s);
    }
}

// -------- final scale: out *= degV (per node) -----------------------------
__global__ void __launch_bounds__(256)
scale_v(v4f* __restrict__ out, const float* __restrict__ degV, int n4) {
    int i = blockIdx.x * blockDim.x + threadIdx.x;
    int stride = gridDim.x * blockDim.x;
    for (; i < n4; i += stride) {
        const int node = i >> 4;                // 16 float4 per 64-float row
        const float s = degV[node];
        v4f t = out[i];
        t.x *= s; t.y *= s; t.z *= s; t.w *= s;
        out[i] = t;
    }
}

extern "C" void kernel_launch(void* const* d_in, const int* in_sizes, int n_in,
                              void* d_out, int out_size, void* d_ws, size_t ws_size,
                              hipStream_t stream) {
    const float* X      = (const float*)d_in[0];   // [N_NODES, IN_CH]
    const int*   vertex = (const int*)  d_in[1];   // [NNZ]
    const int*   edges  = (const int*)  d_in[2];   // [NNZ]
    const float* Wlin   = (const float*)d_in[3];   // [IN_CH, OUT_DIM]
    const float* degE   = (const float*)d_in[4];   // [N_EDGES, 1]
    const float* degV   = (const float*)d_in[5];   // [N_NODES, 1]
    const float* Wedge  = (const float*)d_in[6];   // [N_EDGES, 1]
    float* out = (float*)d_out;                    // [N_NODES, OUT_DIM]

    // Workspace layout (all L2-resident on MI455X: 25.6MB + 5.1MB << 192MB L2)
    float* Xl = (float*)d_ws;                          // [N_NODES, OUT_DIM]
    float* Xe = Xl + (size_t)N_NODES * OUT_DIM;        // [N_EDGES, OUT_DIM]

    const int outN4 = N_NODES * OUT_DIM / 4;
    const int edgN4 = N_EDGES * OUT_DIM / 4;

    // zero the atomic accumulation targets
    zero4_kernel<<<2048, 256, 0, stream>>>((v4f*)out, outN4);
    zero4_kernel<<<2048, 256, 0, stream>>>((v4f*)Xe,  edgN4);

    // Xl = X @ W_lin   (6250 row-block waves / 10 waves per block)
    gemm_wmma_f32<<<625, 320, 0, stream>>>(X, Wlin, Xl);

    // Xe = segment_sum(Xl[vertex], edges)
    scatter_v2e<<<4096, 256, 0, stream>>>(Xl, vertex, edges, Xe);

    // out = segment_sum((Xe*degE*W_edge)[edges], vertex)
    scatter_e2v<<<4096, 256, 0, stream>>>(Xe, vertex, edges, degE, Wedge, out);

    // out *= degV
    scale_v<<<2048, 256, 0, stream>>>((v4f*)out, degV, outN4);
}